// MessageLayer_16088947491015
// MI455X (gfx1250) — compile-verified
//
#include <hip/hip_runtime.h>

typedef __attribute__((ext_vector_type(2))) float v2f;
typedef __attribute__((ext_vector_type(4))) float v4f;
typedef __attribute__((ext_vector_type(8))) float v8f;

// W1s: 16 slabs (row-pairs 2k2,2k2+1 of W1[32][256]) of 256 float2, slab stride
// padded to 272 float2 (544 floats) so lanes 16-31 land on banks 32-63.
#define W1S_SLAB 544
#define W1S_FLOATS (16 * W1S_SLAB)   // 8704
#define W2S_FLOATS 512               // 16 slabs * 16 cols * float2

__global__ __launch_bounds__(256) void msg16_kernel(
    const float* __restrict__ h,   // [E,16]
    const float* __restrict__ e,   // [E,32]
    const float* __restrict__ W1,  // [32,256]
    const float* __restrict__ b1,  // [256]
    const float* __restrict__ W2,  // [32,16]
    const float* __restrict__ b2,  // [16]
    float* __restrict__ out,       // [E,16]
    int E)
{
    __shared__ float smem[W1S_FLOATS + W2S_FLOATS]; // 36864 B
    float* W1s = smem;
    float* W2s = smem + W1S_FLOATS;

    const int tid = threadIdx.x;

    // Stage W1 swizzled: W1s[k2][c] = {W1[2k2][c], W1[2k2+1][c]}, coalesced reads.
    for (int p = tid; p < 16 * 256; p += 256) {
        const int k2 = p >> 8, c = p & 255;
        const float* src = W1 + (2 * k2) * 256 + c;
        W1s[k2 * W1S_SLAB + 2 * c + 0] = src[0];
        W1s[k2 * W1S_SLAB + 2 * c + 1] = src[256];
    }
    // Stage W2 swizzled: W2s[k2][c] = {W2[2k2][c], W2[2k2+1][c]}.
    {
        const int k2 = tid >> 4, c = tid & 15;
        const float* src = W2 + (2 * k2) * 16 + c;
        W2s[tid * 2 + 0] = src[0];
        W2s[tid * 2 + 1] = src[16];
    }
    __syncthreads();

    const int lane  = tid & 31;
    const int m     = lane & 15;   // edge-in-tile (B/N) or row-in-tile (A/M)
    const int halfk = lane >> 4;   // which K pair this lane holds
    const int kb    = 2 * halfk;

    // ---- tile-invariant WMMA operands, hoisted ----
    // sigma GEMM A-operand: A[i,kk] = b1[16*i + 4k + kk]  (contiguous pairs)
    v2f b1A[4];
#pragma unroll
    for (int k = 0; k < 4; ++k)
        b1A[k] = *(const v2f*)(b1 + 16 * m + 4 * k + kb);

    // C init for sigma GEMM: C[i,t] = b2[i], i = r + 8*halfk
    v8f b2C;
#pragma unroll
    for (int r = 0; r < 8; ++r)
        b2C[r] = b2[r + 8 * halfk];

    // W2 GEMM A-operand: A[i,kk] = W2[4k+kk][i], from swizzled LDS
    v2f W2A[8];
#pragma unroll
    for (int k = 0; k < 8; ++k)
        W2A[k] = *(const v2f*)(W2s + (2 * k + halfk) * 32 + 2 * m);

    const float* pA = W1s + halfk * W1S_SLAB + 2 * m; // + 1088*k + 32*n per (k,n)

    const int wavesPerBlk = blockDim.x >> 5;
    const int nWaves = gridDim.x * wavesPerBlk;
    const int numTiles = E >> 4;                 // E % 16 == 0 for this problem
    int w = blockIdx.x * wavesPerBlk + (tid >> 5);

    for (int t = w; t < numTiles; t += nWaves) {
        const int base = t << 4;

        // prefetch next tile's e (2KB) and h (1KB)
        const int tn = t + nWaves;
        if (tn < numTiles) {
            __builtin_prefetch(e + (tn << 4) * 32 + lane * 16, 0, 1);
            __builtin_prefetch(h + (tn << 4) * 16 + lane * 8, 0, 1);
        }

        // B-operand (e^T chunks): B[kk,t] = e[t][4k+kk], contiguous float2
        v2f eB[8];
        const float* ep = e + (base + m) * 32 + kb;
#pragma unroll
        for (int k = 0; k < 8; ++k) eB[k] = *(const v2f*)(ep + 4 * k);

        // B-operand (h^T chunks) for sigma GEMM
        v2f hB[4];
        const float* hp = h + (base + m) * 16 + kb;
#pragma unroll
        for (int k = 0; k < 4; ++k) hB[k] = *(const v2f*)(hp + 4 * k);

        // per-lane h values for the matvec dot: hv[r] = h[base+m][r + 8*halfk]
        const float* hq = h + (base + m) * 16 + 8 * halfk;
        v4f hv0 = *(const v4f*)(hq);
        v4f hv1 = *(const v4f*)(hq + 4);
        float hv[8] = {hv0.x, hv0.y, hv0.z, hv0.w, hv1.x, hv1.y, hv1.z, hv1.w};

        // acc2[i,t] = b2[i] + sum_j b1[16i+j] h[t,j] + sum_d W2[d,i] e[t,d]
        v8f acc2 = b2C;
#pragma unroll
        for (int k = 0; k < 4; ++k)
            acc2 = __builtin_amdgcn_wmma_f32_16x16x4_f32(
                false, b1A[k], false, hB[k], (short)0, acc2, false, false);
#pragma unroll
        for (int k = 0; k < 8; ++k)
            acc2 = __builtin_amdgcn_wmma_f32_16x16x4_f32(
                false, W2A[k], false, eB[k], (short)0, acc2, false, false);

        // main: for each output index n, Gt[16n+.., t] = W1^T tile @ e^T tile,
        // then dot rows against h and reduce the two lane-halves.
        float res[16];
#pragma unroll
        for (int n = 0; n < 16; ++n) {
            v8f acc = {0.f, 0.f, 0.f, 0.f, 0.f, 0.f, 0.f, 0.f};
#pragma unroll
            for (int k = 0; k < 8; ++k) {
                v2f a = *(const v2f*)(pA + k * 1088 + 32 * n);
                acc = __builtin_amdgcn_wmma_f32_16x16x4_f32(
                    false, a, false, eB[k], (short)0, acc, false, false);
            }
            float part = 0.f;
#pragma unroll
            for (int r = 0; r < 8; ++r) part += acc[r] * hv[r];
            part += __shfl_xor(part, 16, 32);   // combine j-halves
            res[n] = part;
        }

        // fold acc2 (lane holds i = r + 8*halfk for edge m) into res
#pragma unroll
        for (int r = 0; r < 8; ++r) {
            const float own = acc2[r];
            const float oth = __shfl_xor(own, 16, 32);
            res[r]     += halfk ? oth : own;
            res[r + 8] += halfk ? own : oth;
        }

        // lanes 0-15 each own one edge row; coalesced float4 stores
        if (lane < 16) {
            float* op = out + (base + m) * 16;
#pragma unroll
            for (int q = 0; q < 4; ++q) {
                v4f v = {res[4 * q], res[4 * q + 1], res[4 * q + 2], res[4 * q + 3]};
                *(v4f*)(op + 4 * q) = v;
            }
        }
    }
}

extern "C" void kernel_launch(void* const* d_in, const int* in_sizes, int n_in,
                              void* d_out, int out_size, void* d_ws, size_t ws_size,
                              hipStream_t stream) {
    const float* h  = (const float*)d_in[0];
    const float* e  = (const float*)d_in[1];
    const float* W1 = (const float*)d_in[2];
    const float* b1 = (const float*)d_in[3];
    const float* W2 = (const float*)d_in[4];
    const float* b2 = (const float*)d_in[5];
    float* out = (float*)d_out;

    const int E = in_sizes[0] / 16;       // h is [E,16]
    const int numTiles = E / 16;          // E = 1.6M -> 100000 tiles
    int blocks = 1024;                    // 8192 waves grid-stride; amortize W1 LDS staging
    const int maxBlocks = (numTiles + 7) / 8;
    if (blocks > maxBlocks) blocks = maxBlocks;
    if (blocks < 1) blocks = 1;

    msg16_kernel<<<blocks, 256, 0, stream>>>(h, e, W1, b1, W2, b2, out, E);
}